// CrossCosineEmbeddingLoss_26525718020130
// MI455X (gfx1250) — compile-verified
//
#include <hip/hip_runtime.h>
#include <math.h>

typedef __attribute__((ext_vector_type(2))) float v2f;
typedef __attribute__((ext_vector_type(8))) float v8f;

#define N_ROWS 8192
#define DIM 128
#define TILE 128
#define LSTRIDE 132   // 128 + 4 pad: conflict-free 64-bank spread for b64 frag loads

__global__ void init_acc_kernel(double* acc) {
    if (threadIdx.x == 0 && blockIdx.x == 0) *acc = 0.0;
}

__global__ __launch_bounds__(256) void row_norms_kernel(const float* __restrict__ x,
                                                        const float* __restrict__ y,
                                                        float* __restrict__ nx,
                                                        float* __restrict__ ny) {
    const int wave = threadIdx.x >> 5;
    const int lane = threadIdx.x & 31;
    const int r = blockIdx.x * 8 + wave;            // 0..16383
    const float* src;
    float* dst;
    int row;
    if (r < N_ROWS) { src = x; dst = nx; row = r; }
    else            { src = y; dst = ny; row = r - N_ROWS; }
    const float4* p = (const float4*)(src + (size_t)row * DIM);
    float4 v = p[lane];                              // 32 lanes * 4 floats = 128
    float s = v.x * v.x + v.y * v.y + v.z * v.z + v.w * v.w;
    #pragma unroll
    for (int off = 16; off > 0; off >>= 1) s += __shfl_xor(s, off, 32);
    if (lane == 0) dst[row] = sqrtf(s);
}

__global__ __launch_bounds__(256) void cosloss_gemm_kernel(const float* __restrict__ x,
                                                           const float* __restrict__ y,
                                                           const float* __restrict__ nx,
                                                           const float* __restrict__ ny,
                                                           double* __restrict__ acc) {
    extern __shared__ float smem[];
    float* lx   = smem;                      // TILE * LSTRIDE
    float* ly   = lx + TILE * LSTRIDE;       // TILE * LSTRIDE
    float* lnx  = ly + TILE * LSTRIDE;       // TILE
    float* lny  = lnx + TILE;                // TILE
    float* wsum = lny + TILE;                // 8

    const int tid   = threadIdx.x;
    const int tileM = blockIdx.x * TILE;
    const int tileN = blockIdx.y * TILE;

    // ---- cooperative stage of both tiles (full K=128) into LDS ----
    {
        const float4* gx = (const float4*)(x + (size_t)tileM * DIM);
        const float4* gy = (const float4*)(y + (size_t)tileN * DIM);
        const int f4pr = DIM / 4;            // 32 float4 per row
        #pragma unroll
        for (int i = tid; i < TILE * f4pr; i += 256) {
            const int row = i / f4pr;
            const int c4  = i % f4pr;
            float4 vx = gx[(size_t)row * f4pr + c4];
            float4 vy = gy[(size_t)row * f4pr + c4];
            float* dx = &lx[row * LSTRIDE + c4 * 4];
            dx[0] = vx.x; dx[1] = vx.y; dx[2] = vx.z; dx[3] = vx.w;
            float* dy = &ly[row * LSTRIDE + c4 * 4];
            dy[0] = vy.x; dy[1] = vy.y; dy[2] = vy.z; dy[3] = vy.w;
        }
        if (tid < TILE) {
            lnx[tid] = nx[tileM + tid];
            lny[tid] = ny[tileN + tid];
        }
    }
    __syncthreads();

    // ---- per-wave 32x64 sub-tile: 2 (M) x 4 (N) accumulators of 16x16 ----
    const int lane = tid & 31;
    const int wave = tid >> 5;
    const int m0 = (wave >> 1) * 32;         // wave row offset in tile
    const int n0 = (wave & 1) * 64;          // wave col offset in tile
    const int fr = lane & 15;                // fragment M (A) / N (B) index
    const int kb = (lane >> 4) << 1;         // K sub-offset: 0 or 2 (ISA 16x4 layout)

    v8f acc_t[2][4];
    #pragma unroll
    for (int mt = 0; mt < 2; ++mt)
        #pragma unroll
        for (int nt = 0; nt < 4; ++nt)
            acc_t[mt][nt] = (v8f){};

    for (int k = 0; k < DIM; k += 4) {
        v2f a[2], b[4];
        #pragma unroll
        for (int mt = 0; mt < 2; ++mt)
            a[mt] = *(const v2f*)&lx[(m0 + mt * 16 + fr) * LSTRIDE + k + kb];
        #pragma unroll
        for (int nt = 0; nt < 4; ++nt)
            b[nt] = *(const v2f*)&ly[(n0 + nt * 16 + fr) * LSTRIDE + k + kb];
        #pragma unroll
        for (int mt = 0; mt < 2; ++mt)
            #pragma unroll
            for (int nt = 0; nt < 4; ++nt)
                acc_t[mt][nt] = __builtin_amdgcn_wmma_f32_16x16x4_f32(
                    false, a[mt], false, b[nt], (short)0, acc_t[mt][nt], false, false);
    }

    // ---- fused epilogue: cosine + CosineEmbeddingLoss + reduction ----
    // C layout: VGPR r -> M = r (lanes 0-15) or r+8 (lanes 16-31); N = lane%16
    float lsum = 0.0f;
    const int hlf = (lane >> 4) << 3;        // 0 or 8
    #pragma unroll
    for (int mt = 0; mt < 2; ++mt) {
        #pragma unroll
        for (int nt = 0; nt < 4; ++nt) {
            const int lj  = n0 + nt * 16 + fr;
            const int gj  = tileN + lj;
            const float nyv = lny[lj];
            #pragma unroll
            for (int r = 0; r < 8; ++r) {
                const int li  = m0 + mt * 16 + r + hlf;
                const int gi  = tileM + li;
                const float nxv = lnx[li];
                const float sim = acc_t[mt][nt][r] / fmaxf(nxv * nyv, 1e-8f);
                lsum += (gi == gj) ? (1.0f - sim) : fmaxf(sim, 0.0f);
            }
        }
    }
    #pragma unroll
    for (int off = 16; off > 0; off >>= 1) lsum += __shfl_xor(lsum, off, 32);
    if (lane == 0) wsum[wave] = lsum;
    __syncthreads();
    if (tid == 0) {
        float bs = 0.0f;
        #pragma unroll
        for (int w = 0; w < 8; ++w) bs += wsum[w];
        atomicAdd(acc, (double)bs);          // global_atomic_add_f64, one per block
    }
}

__global__ void finalize_kernel(const double* __restrict__ acc, float* __restrict__ out) {
    if (threadIdx.x == 0 && blockIdx.x == 0)
        out[0] = (float)(*acc / ((double)N_ROWS * (double)N_ROWS));
}

extern "C" void kernel_launch(void* const* d_in, const int* in_sizes, int n_in,
                              void* d_out, int out_size, void* d_ws, size_t ws_size,
                              hipStream_t stream) {
    const float* x = (const float*)d_in[0];
    const float* y = (const float*)d_in[1];
    float* out = (float*)d_out;

    // workspace layout: [double acc][pad to 64B][nx: 8192 f32][ny: 8192 f32]
    double* acc = (double*)d_ws;
    float* nx = (float*)((char*)d_ws + 64);
    float* ny = nx + N_ROWS;

    init_acc_kernel<<<1, 1, 0, stream>>>(acc);
    row_norms_kernel<<<(2 * N_ROWS) / 8, 256, 0, stream>>>(x, y, nx, ny);

    dim3 grid(N_ROWS / TILE, N_ROWS / TILE);   // 64 x 64
    size_t shmem = (size_t)(2 * TILE * LSTRIDE + 2 * TILE + 8) * sizeof(float);
    cosloss_gemm_kernel<<<grid, 256, shmem, stream>>>(x, y, nx, ny, acc);

    finalize_kernel<<<1, 1, 0, stream>>>(acc, out);
}